// AutoCorrelation_14705968021882
// MI455X (gfx1250) — compile-verified
//
#include <hip/hip_runtime.h>
#include <math.h>

typedef __attribute__((ext_vector_type(2))) float v2f;
typedef __attribute__((ext_vector_type(8))) float v8f;

#define L_SEQ 2048
#define DMODEL 512
#define TOPK 8

// ---------------------------------------------------------------------------
// Generic fp32 WMMA GEMM: C[m,n] = sum_k A[m,k]*W[k,n] + bias[n]
//   M = 16384 (B*L), N = K = 512.
//   AT   : A physically stored transposed-per-batch: Aphys[(b*512+k)*2048 + t],
//          with m = b*2048 + t  (used for the output projection reading dagg).
//   OUTT : store output in head-transposed, t-contiguous layout
//          out[(b*512+n)*2048 + t]  (used for Q/K projections feeding autocorr).
// Block tile 64x64, 256 threads = 8 waves (4 in M x 2 in N), each wave does a
// 16x32 slab = 2 accumulators, K-chunks of 32 staged in LDS, 16x16x4 f32 WMMA.
// ---------------------------------------------------------------------------
template <bool AT, bool OUTT>
__global__ __launch_bounds__(256) void gemm_wmma_f32(
    const float* __restrict__ A, const float* __restrict__ W,
    const float* __restrict__ bias, float* __restrict__ out) {
  __shared__ float As[64][33];
  __shared__ float Bs[32][65];
  __shared__ float Cs[64][65];  // only used on OUTT path

  const int tid  = threadIdx.x;
  const int lane = tid & 31;
  const int wave = tid >> 5;
  const int wm   = wave & 3;   // wave row   (16 m-rows each)
  const int wn   = wave >> 2;  // wave col   (32 n-cols each)
  const int m0   = blockIdx.x * 64;
  const int n0   = blockIdx.y * 64;
  const int half = lane >> 4;  // lanes 16-31 hold K=2,3 (per ISA A layout)
  const int l16  = lane & 15;

  v8f acc0 = {};
  v8f acc1 = {};

  for (int k0 = 0; k0 < DMODEL; k0 += 32) {
    __syncthreads();
    if (!AT) {
      #pragma unroll
      for (int i = 0; i < 2; ++i) {  // 512 float4 = 64x32 A tile
        int f = tid + i * 256;
        int row = f >> 3;
        int c4  = (f & 7) << 2;
        float4 av = *(const float4*)(A + (size_t)(m0 + row) * DMODEL + k0 + c4);
        As[row][c4 + 0] = av.x; As[row][c4 + 1] = av.y;
        As[row][c4 + 2] = av.z; As[row][c4 + 3] = av.w;
      }
    } else {
      const int bb = m0 >> 11;    // batch index  (m0 / 2048)
      const int t0 = m0 & 2047;
      #pragma unroll
      for (int i = 0; i < 8; ++i) {  // 2048 scalars, contiguous along t
        int f = tid + i * 256;
        int kcol = f >> 6;
        int mrow = f & 63;
        As[mrow][kcol] =
            A[(size_t)(bb * DMODEL + k0 + kcol) * L_SEQ + t0 + mrow];
      }
    }
    #pragma unroll
    for (int i = 0; i < 2; ++i) {  // 512 float4 = 32x64 B tile
      int f = tid + i * 256;
      int row = f >> 4;
      int c4  = (f & 15) << 2;
      float4 bv = *(const float4*)(W + (size_t)(k0 + row) * DMODEL + n0 + c4);
      Bs[row][c4 + 0] = bv.x; Bs[row][c4 + 1] = bv.y;
      Bs[row][c4 + 2] = bv.z; Bs[row][c4 + 3] = bv.w;
    }
    __syncthreads();

    #pragma unroll
    for (int kk = 0; kk < 32; kk += 4) {
      v2f a, b0, b1;
      a.x  = As[wm * 16 + l16][kk + half * 2 + 0];
      a.y  = As[wm * 16 + l16][kk + half * 2 + 1];
      b0.x = Bs[kk + half * 2 + 0][wn * 32 + l16];
      b0.y = Bs[kk + half * 2 + 1][wn * 32 + l16];
      b1.x = Bs[kk + half * 2 + 0][wn * 32 + 16 + l16];
      b1.y = Bs[kk + half * 2 + 1][wn * 32 + 16 + l16];
      acc0 = __builtin_amdgcn_wmma_f32_16x16x4_f32(false, a, false, b0,
                                                   (short)0, acc0, false, false);
      acc1 = __builtin_amdgcn_wmma_f32_16x16x4_f32(false, a, false, b1,
                                                   (short)0, acc1, false, false);
    }
  }

  if (!OUTT) {
    #pragma unroll
    for (int r = 0; r < 8; ++r) {
      int m = m0 + wm * 16 + r + half * 8;
      int n = n0 + wn * 32 + l16;
      out[(size_t)m * DMODEL + n]      = acc0[r] + bias[n];
      out[(size_t)m * DMODEL + n + 16] = acc1[r] + bias[n + 16];
    }
  } else {
    __syncthreads();
    #pragma unroll
    for (int r = 0; r < 8; ++r) {
      int mr = wm * 16 + r + half * 8;
      Cs[mr][wn * 32 + l16]      = acc0[r];
      Cs[mr][wn * 32 + 16 + l16] = acc1[r];
    }
    __syncthreads();
    const int bb = m0 >> 11;
    const int t0 = m0 & 2047;
    for (int i = 0; i < 16; ++i) {  // 4096 outputs, t-contiguous stores
      int f = tid + i * 256;
      int n  = f >> 6;
      int tt = f & 63;
      out[(size_t)(bb * DMODEL + n0 + n) * L_SEQ + t0 + tt] =
          Cs[tt][n] + bias[n0 + n];
    }
  }
}

// ---------------------------------------------------------------------------
// AutoCorrelation per (b,h,d) row via Toeplitz-factored WMMA:
//   R[tb+16a+n] = sum_s q[(s+tb+16a)&2047] * k[(s-n)&2047]
// 8 waves x 256 lags = 2048 lags per block. Then top-8 (+softmax) and the
// delayed-q aggregation. qs duplicated (4096) / ks shifted+duplicated so the
// inner WMMA loop needs no modulo.
// ---------------------------------------------------------------------------
__global__ __launch_bounds__(256) void autocorr_topk(
    const float* __restrict__ qt, const float* __restrict__ kt,
    float* __restrict__ dagg) {
  __shared__ float qs[4096];
  __shared__ float ks[4096];  // ks[i] = k[(i-16) & 2047]
  __shared__ float R[2048];
  __shared__ float rv[256];
  __shared__ int   ri[256];
  __shared__ float topv[TOPK];
  __shared__ int   topi[TOPK];

  const int row  = blockIdx.x;  // b*512 + h*64 + d
  const int tid  = threadIdx.x;
  const int lane = tid & 31;
  const int wave = tid >> 5;
  const int half = lane >> 4;
  const int l16  = lane & 15;

  const float* qrow = qt + (size_t)row * L_SEQ;
  const float* krow = kt + (size_t)row * L_SEQ;

  for (int i = tid; i < 4096; i += 256) {
    qs[i] = qrow[i & 2047];
    ks[i] = krow[(i - 16) & 2047];
  }
  __syncthreads();

  const int tb   = wave << 8;                  // lag-tile base (256 lags/wave)
  const int aoff = tb + 16 * l16 + 2 * half;   // q index offset (a = l16)
  const int boff = 16 - l16 + 2 * half;        // k index offset (n = l16)

  v8f acc = {};
  #pragma unroll 8
  for (int s0 = 0; s0 < L_SEQ; s0 += 4) {
    v2f a, b;
    a.x = qs[s0 + aoff];
    a.y = qs[s0 + aoff + 1];
    b.x = ks[s0 + boff];
    b.y = ks[s0 + boff + 1];
    acc = __builtin_amdgcn_wmma_f32_16x16x4_f32(false, a, false, b, (short)0,
                                                acc, false, false);
  }

  #pragma unroll
  for (int r = 0; r < 8; ++r) {
    int a = r + half * 8;
    R[tb + 16 * a + l16] = acc[r];
  }
  __syncthreads();

  // top-8 with lax.top_k tie-breaking (lower index wins on equal values)
  for (int j = 0; j < TOPK; ++j) {
    float bv = -INFINITY;
    int   bi = 0;
    #pragma unroll
    for (int u = 0; u < 8; ++u) {
      int idx = tid * 8 + u;
      float v = R[idx];
      if (v > bv) { bv = v; bi = idx; }
    }
    rv[tid] = bv;
    ri[tid] = bi;
    __syncthreads();
    for (int s = 128; s > 0; s >>= 1) {
      if (tid < s) {
        float ov = rv[tid + s];
        int   oi = ri[tid + s];
        if (ov > rv[tid] || (ov == rv[tid] && oi < ri[tid])) {
          rv[tid] = ov;
          ri[tid] = oi;
        }
      }
      __syncthreads();
    }
    if (tid == 0) {
      topv[j] = rv[0];
      topi[j] = ri[0];
      R[ri[0]] = -INFINITY;  // mask winner
    }
    __syncthreads();
  }

  // softmax over the 8 weights (redundant per-thread) + delayed aggregation
  float w[TOPK];
  int   dly[TOPK];
  float mx = -INFINITY;
  #pragma unroll
  for (int j = 0; j < TOPK; ++j) {
    w[j]   = topv[j];
    dly[j] = topi[j];
    mx = fmaxf(mx, w[j]);
  }
  float sum = 0.f;
  #pragma unroll
  for (int j = 0; j < TOPK; ++j) {
    w[j] = __expf(w[j] - mx);
    sum += w[j];
  }
  const float inv = 1.0f / sum;

  for (int t = tid; t < L_SEQ; t += 256) {
    float o = 0.f;
    #pragma unroll
    for (int j = 0; j < TOPK; ++j) o += w[j] * qs[t + dly[j]];
    // t-contiguous layout: dagg[(b*512+n)*2048 + t], row == b*512+n
    dagg[(size_t)row * L_SEQ + t] = o * inv;
  }
}

// ---------------------------------------------------------------------------
extern "C" void kernel_launch(void* const* d_in, const int* in_sizes, int n_in,
                              void* d_out, int out_size, void* d_ws,
                              size_t ws_size, hipStream_t stream) {
  (void)in_sizes; (void)n_in; (void)out_size; (void)ws_size;
  const float* q  = (const float*)d_in[0];
  const float* k  = (const float*)d_in[1];
  // d_in[2] (v), d_in[7] (Wv), d_in[8] (bv) are dead: the reference gathers
  // its aggregation pattern from q_t, so vh never reaches the output.
  const float* Wq = (const float*)d_in[3];
  const float* bq = (const float*)d_in[4];
  const float* Wk = (const float*)d_in[5];
  const float* bk = (const float*)d_in[6];
  const float* Wo = (const float*)d_in[9];
  const float* bo = (const float*)d_in[10];
  float* out = (float*)d_out;

  const size_t mat = (size_t)16384 * DMODEL;  // 32 MB each
  float* qt   = (float*)d_ws;
  float* kt   = qt + mat;
  float* dagg = kt + mat;

  dim3 grid(16384 / 64, DMODEL / 64);  // 256 x 8
  dim3 blk(256);
  gemm_wmma_f32<false, true><<<grid, blk, 0, stream>>>(q, Wq, bq, qt);
  gemm_wmma_f32<false, true><<<grid, blk, 0, stream>>>(k, Wk, bk, kt);
  autocorr_topk<<<dim3(8 * 512), blk, 0, stream>>>(qt, kt, dagg);
  gemm_wmma_f32<true, false><<<grid, blk, 0, stream>>>(dagg, Wo, bo, out);
}